// GAT_36009005809883
// MI455X (gfx1250) — compile-verified
//
#include <hip/hip_runtime.h>
#include <math.h>

// GAT layer on gfx1250:
//   K1: Wh = h @ W + bW  via V_WMMA_F32_16X16X4_F32  (f32 matrix path, wave32)
//   K2: s_src/s_dst per (node, head)
//   K3: sorted-dst segment online-softmax + weighted aggregation (1 wave / (node,head))

#define ALPHA 0.2f

typedef __attribute__((ext_vector_type(2))) float v2f;
typedef __attribute__((ext_vector_type(8))) float v8f;

constexpr int N_NODES = 50000;
constexpr int N_EDGES = 800000;
constexpr int F_IN    = 256;
constexpr int NH      = 4;     // heads
constexpr int HD      = 32;    // dim per head
constexpr int C_OUT   = NH * HD;   // 128

// ---------------------------------------------------------------------------
// Kernel 1: per-head projection GEMM, Wh[n, h*32+d] = sum_i h[n,i]*W[h,i,d] + bW[h,d]
// Block = 256 threads (8 waves). Block b owns rows [16b,16b+16); wave w owns
// cols [16w,16w+16). f32 WMMA 16x16x4, K swept 256/4 = 64 steps.
// Fragment layout (ISA 7.12.2, f32):
//   A 16x4 : lanes 0-15 -> M=lane, k-half 0; lanes 16-31 -> M=lane-16, k-half 2
//            v0 = A[M][k0], v1 = A[M][k0+1]   (k0 = k + 2*(lane>>4))
//   B 4x16 : lanes 0-15 -> N=lane, rows {k, k+1}; lanes 16-31 -> rows {k+2, k+3}
//   C/D    : vgpr r, lane L -> M = r + 8*(L>>4), N = L&15
// ---------------------------------------------------------------------------
__global__ __launch_bounds__(256) void gat_gemm_wmma(
    const float* __restrict__ h, const float* __restrict__ W,
    const float* __restrict__ bW, float* __restrict__ Wh)
{
  const int lane  = threadIdx.x & 31;
  const int wave  = threadIdx.x >> 5;         // 0..7 -> N tile
  const int tileM = blockIdx.x * 16;
  const int col   = wave * 16 + (lane & 15);  // global output column 0..127
  const int head  = col >> 5;
  const int dcol  = col & 31;
  const int khalf = (lane >> 4) << 1;         // 0 or 2

  // A: row (tileM + lane&15), starting at k-offset khalf (8B aligned: k%4==0)
  const float* arow = h + (size_t)(tileM + (lane & 15)) * F_IN + khalf;
  // B: W[head][k+khalf][dcol]; consecutive k rows are 32 floats apart
  const float* bcol = W + (size_t)head * F_IN * HD + (size_t)khalf * HD + dcol;

  v8f c = {};
  #pragma unroll 4
  for (int k = 0; k < F_IN; k += 4) {
    v2f a = *(const v2f*)(arow + k);
    v2f b;
    b.x = bcol[(size_t)k * HD];
    b.y = bcol[(size_t)k * HD + HD];
    c = __builtin_amdgcn_wmma_f32_16x16x4_f32(
        /*neg_a=*/false, a, /*neg_b=*/false, b,
        /*c_mod=*/(short)0, c, /*reuse_a=*/false, /*reuse_b=*/false);
  }

  const float bias  = bW[col];
  const int rowBase = tileM + ((lane >> 4) << 3);
  #pragma unroll
  for (int r = 0; r < 8; ++r) {
    Wh[(size_t)(rowBase + r) * C_OUT + col] = c[r] + bias;
  }
}

// ---------------------------------------------------------------------------
// Kernel 2: attention score halves per (node, head)
//   s_src[n,h] = Wh[n,h,:] . a[h,:D]      s_dst[n,h] = Wh[n,h,:] . a[h,D:]
// One thread per (n,h); tid*32 indexes Wh directly since C_OUT = NH*HD.
// ---------------------------------------------------------------------------
__global__ __launch_bounds__(256) void gat_scores(
    const float* __restrict__ Wh, const float* __restrict__ a,
    float* __restrict__ s_src, float* __restrict__ s_dst)
{
  const int tid = blockIdx.x * blockDim.x + threadIdx.x;
  if (tid >= N_NODES * NH) return;
  const int head = tid & (NH - 1);

  const float4* w  = (const float4*)(Wh + (size_t)tid * HD);
  const float4* as = (const float4*)(a + head * 2 * HD);
  const float4* ad = (const float4*)(a + head * 2 * HD + HD);

  float ss = 0.f, sd = 0.f;
  #pragma unroll
  for (int i = 0; i < HD / 4; ++i) {
    float4 wv = w[i], av = as[i], dv = ad[i];
    ss += wv.x * av.x + wv.y * av.y + wv.z * av.z + wv.w * av.w;
    sd += wv.x * dv.x + wv.y * dv.y + wv.z * dv.z + wv.w * dv.w;
  }
  s_src[tid] = ss;
  s_dst[tid] = sd;
}

// ---------------------------------------------------------------------------
// Kernel 3: segment softmax + aggregation. dst is SORTED, so each node owns a
// contiguous edge range, found by binary search. One wave32 per (node, head);
// lanes map to the D=32 output channels. Online softmax: score is uniform
// across lanes (broadcast loads), max-rescale keeps it single-pass.
// ---------------------------------------------------------------------------
__global__ __launch_bounds__(256) void gat_aggregate(
    const float* __restrict__ Wh, const float* __restrict__ s_src,
    const float* __restrict__ s_dst, const float* __restrict__ ba,
    const int* __restrict__ src, const int* __restrict__ dst,
    float* __restrict__ out)
{
  const int lane = threadIdx.x & 31;
  const int wid  = blockIdx.x * (blockDim.x >> 5) + (threadIdx.x >> 5);
  const int n    = wid >> 2;           // node
  const int head = wid & (NH - 1);

  // lower_bound(dst, n) and lower_bound(dst, n+1) — uniform across the wave
  int lo, hi;
  {
    int l = 0, r = N_EDGES;
    while (l < r) { int m = (l + r) >> 1; if (dst[m] < n) l = m + 1; else r = m; }
    lo = l;
    r = N_EDGES;
    while (l < r) { int m = (l + r) >> 1; if (dst[m] <= n) l = m + 1; else r = m; }
    hi = l;
  }

  const float sdn = s_dst[n * NH + head] + ba[head];

  float m = -INFINITY, den = 0.f, acc = 0.f;
  for (int e = lo; e < hi; ++e) {
    const int s = src[e];
    float sc = s_src[s * NH + head] + sdn;
    sc = sc > 0.f ? sc : ALPHA * sc;            // LeakyReLU
    if (sc > m) {                                // wave-uniform branch
      const float corr = __expf(m - sc);         // first iter: exp(-inf)=0
      den *= corr;
      acc *= corr;
      m = sc;
    }
    const float p = __expf(sc - m);
    den += p;
    acc += p * Wh[(size_t)s * C_OUT + head * HD + lane];  // coalesced 128B row
  }
  // empty segment: den==0 -> 0, matching reference's finite-guard + 1e-12 clamp
  out[(size_t)n * C_OUT + head * HD + lane] = acc / fmaxf(den, 1e-12f);
}

// ---------------------------------------------------------------------------
extern "C" void kernel_launch(void* const* d_in, const int* in_sizes, int n_in,
                              void* d_out, int out_size, void* d_ws, size_t ws_size,
                              hipStream_t stream) {
  (void)in_sizes; (void)n_in; (void)out_size; (void)ws_size;
  const float* h   = (const float*)d_in[0];   // [N, 256]
  const float* W   = (const float*)d_in[1];   // [4, 256, 32]
  const float* bW  = (const float*)d_in[2];   // [4, 32]
  const float* a   = (const float*)d_in[3];   // [4, 64]
  const float* ba  = (const float*)d_in[4];   // [4]
  const int*   src = (const int*)d_in[5];     // [E]
  const int*   dst = (const int*)d_in[6];     // [E] sorted
  float* out = (float*)d_out;                 // [N, 128]

  float* Wh    = (float*)d_ws;                       // N*128 floats = 25.6 MB
  float* s_src = Wh + (size_t)N_NODES * C_OUT;       // N*4
  float* s_dst = s_src + (size_t)N_NODES * NH;       // N*4

  gat_gemm_wmma<<<N_NODES / 16, 256, 0, stream>>>(h, W, bW, Wh);
  gat_scores<<<(N_NODES * NH + 255) / 256, 256, 0, stream>>>(Wh, a, s_src, s_dst);
  gat_aggregate<<<(N_NODES * NH) / 8, 256, 0, stream>>>(Wh, s_src, s_dst, ba,
                                                        src, dst, out);
}